// Canny_78022375899468
// MI455X (gfx1250) — compile-verified
//
#include <hip/hip_runtime.h>
#include <math.h>

// ---------------------------------------------------------------------------
// Canny edge detector, fused, for (16,3,1024,1024) fp32 -> (16,1,1024,1024).
// Pass A: grey = sqrt(r^2+g^2+b^2); global max (atomicMax on float bits,
//         valid since grey >= 0); optionally caches grey plane in d_ws.
// Pass B: per-32x32-tile fused pipeline entirely in LDS:
//   async-DMA grey (or 3 channel) tiles with reflect-mapped halo-6 ->
//   normalize -> gauss 5x5 -> sobel 5x5 -> mag/phase -> NMS + low thresh ->
//   hysteresis -> out.
// CDNA5 paths: global_load_async_to_lds_b32 + s_wait_asynccnt (ASYNCcnt),
// wave32 sizing (256 thr = 8 waves), 320KB-LDS-friendly tiling.
// Memory-bound problem (AI ~5 FLOP/B): no matrix contraction, so WMMA cannot
// reduce bytes; async-DMA + L2 halo reuse is the right CDNA5 path.
// ---------------------------------------------------------------------------

#define TSZ  32            // output tile
#define GE   (TSZ + 12)    // grey extent   (halo 6 = 2 gauss + 2 sobel + 1 NMS + 1 hyst)
#define BE   (TSZ + 8)     // blurred extent(halo 4)
#define ME   (TSZ + 4)     // mag extent    (halo 2)
#define NE   (TSZ + 2)     // nms extent    (halo 1)
#define NPIX (1024 * 1024)

__device__ __forceinline__ int reflect1024(int i) {
  i = (i < 0) ? -i : i;                 // jnp.pad 'reflect': -1 -> 1
  return (i > 1023) ? (2046 - i) : i;   // 1024 -> 1022
}

// Async DMA: copies 4 bytes per lane from global address gaddr to LDS byte offset lds_byte_off. ASYNCcnt-tracked.
__device__ __forceinline__ void async_ld_f32(unsigned lds_byte_off, const float* gaddr) {
  asm volatile("global_load_async_to_lds_b32 %0, %1, off"
               :: "v"(lds_byte_off), "v"(gaddr) : "memory");
}
__device__ __forceinline__ void wait_async0() {
  asm volatile("s_wait_asynccnt 0" ::: "memory");
}
__device__ __forceinline__ unsigned lds_off_of(const void* p) {
  // Flat LDS aperture keeps the LDS byte offset in the low 32 bits.
  return (unsigned)(unsigned long long)p;
}

// ------------------------------- Pass A ------------------------------------
template <bool STORE_GREY>
__global__ __launch_bounds__(256) void canny_max_kernel(
    const float* __restrict__ in, float* __restrict__ grey,
    unsigned int* __restrict__ gmax) {
  __shared__ float red[256];
  float m = 0.0f;
  const int stride = gridDim.x * blockDim.x;
  for (int p = blockIdx.x * blockDim.x + threadIdx.x; p < 16 * NPIX; p += stride) {
    const int img = p >> 20;
    const int pix = p & (NPIX - 1);
    const size_t base = ((size_t)img * 3) << 20;
    const float r = in[base + pix];
    const float g = in[base + NPIX + pix];
    const float b = in[base + 2 * (size_t)NPIX + pix];
    const float v = sqrtf(r * r + g * g + b * b);
    if (STORE_GREY) grey[p] = v;            // unnormalized grey plane cache
    m = fmaxf(m, v);
  }
  red[threadIdx.x] = m;
  __syncthreads();
  for (int s = 128; s > 0; s >>= 1) {
    if (threadIdx.x < s) red[threadIdx.x] = fmaxf(red[threadIdx.x], red[threadIdx.x + s]);
    __syncthreads();
  }
  if (threadIdx.x == 0) atomicMax(gmax, __float_as_uint(red[0]));
}

// ------------------------------- Pass B ------------------------------------
template <bool USE_GREY>
__global__ __launch_bounds__(256) void canny_main_kernel(
    const float* __restrict__ in, const float* __restrict__ grey,
    float* __restrict__ out, const unsigned int* __restrict__ gmaxp) {
  // Gauss 5x5, sigma=0.1 over linspace(-1,1,5), normalized (precomputed).
  constexpr float g0   = 0.99998509355f;   // r2=0
  constexpr float g025 = 3.7265976e-06f;   // r2=0.25
  constexpr float g05  = 1.3887737e-11f;   // r2=0.5
  constexpr float g1   = 1.9287211e-22f;   // r2=1.0
  constexpr float g125 = 7.1876753e-28f;   // r2=1.25
  constexpr float g2   = 3.7200205e-44f;   // r2=2.0 (subnormal, negligible)
  const float GW[5][5] = {
      {g2,   g125, g1,   g125, g2  },
      {g125, g05,  g025, g05,  g125},
      {g1,   g025, g0,   g025, g1  },
      {g125, g05,  g025, g05,  g125},
      {g2,   g125, g1,   g125, g2  }};
  // Sobel s[i][j] = x/(x^2+y^2)/6 with denom[:,2]=1, x=lin[i], y=lin[j].
  const float SW[5][5] = {
      {-2.f/8.f/6.f, -2.f/5.f/6.f, -2.f/1.f/6.f, -2.f/5.f/6.f, -2.f/8.f/6.f},
      {-1.f/5.f/6.f, -1.f/2.f/6.f, -1.f/1.f/6.f, -1.f/2.f/6.f, -1.f/5.f/6.f},
      { 0.f,          0.f,          0.f,          0.f,          0.f        },
      { 1.f/5.f/6.f,  1.f/2.f/6.f,  1.f/1.f/6.f,  1.f/2.f/6.f,  1.f/5.f/6.f},
      { 2.f/8.f/6.f,  2.f/5.f/6.f,  2.f/1.f/6.f,  2.f/5.f/6.f,  2.f/8.f/6.f}};

  __shared__ float s_g[GE * GE];     // grey tile (normalized after stage 2)
  __shared__ float s_blur[BE * BE];
  __shared__ float s_mag[ME * ME];
  __shared__ int   s_ph[ME * ME];
  __shared__ float s_nms[NE * NE];

  const int tid = threadIdx.x;
  const int n   = blockIdx.z;
  const int ty0 = blockIdx.y * TSZ;
  const int tx0 = blockIdx.x * TSZ;
  const float gmax = __uint_as_float(*gmaxp);

  // Stage 1+2: async DMA the tile (reflect-mapped halo-6) into LDS, normalize.
  if constexpr (USE_GREY) {
    const size_t ibase = (size_t)n << 20;
    for (int idx = tid; idx < GE * GE; idx += 256) {
      const int ly = idx / GE, lx = idx - ly * GE;
      const int ry = reflect1024(ty0 - 6 + ly);
      const int rx = reflect1024(tx0 - 6 + lx);
      async_ld_f32(lds_off_of(&s_g[idx]), grey + ibase + ((size_t)ry << 10) + rx);
    }
    wait_async0();
    __syncthreads();
    for (int idx = tid; idx < GE * GE; idx += 256) s_g[idx] = s_g[idx] / gmax;
    __syncthreads();
  } else {
    __shared__ float s_c1[GE * GE];
    __shared__ float s_c2[GE * GE];
    const size_t cbase = ((size_t)n * 3) << 20;
    for (int idx = tid; idx < GE * GE; idx += 256) {
      const int ly = idx / GE, lx = idx - ly * GE;
      const int ry = reflect1024(ty0 - 6 + ly);
      const int rx = reflect1024(tx0 - 6 + lx);
      const float* p0 = in + cbase + ((size_t)ry << 10) + rx;
      async_ld_f32(lds_off_of(&s_g[idx]),  p0);
      async_ld_f32(lds_off_of(&s_c1[idx]), p0 + NPIX);
      async_ld_f32(lds_off_of(&s_c2[idx]), p0 + 2 * NPIX);
    }
    wait_async0();
    __syncthreads();
    for (int idx = tid; idx < GE * GE; idx += 256) {
      const float r = s_g[idx], g = s_c1[idx], b = s_c2[idx];
      s_g[idx] = sqrtf(r * r + g * g + b * b) / gmax;
    }
    __syncthreads();
  }

  // Stage 3: 5x5 gaussian, reflect-padded. Slot(c) holds blur(reflect(c)).
  for (int idx = tid; idx < BE * BE; idx += 256) {
    const int ly = idx / BE, lx = idx - ly * BE;
    const int by = reflect1024(ty0 - 4 + ly) - (ty0 - 6);
    const int bx = reflect1024(tx0 - 4 + lx) - (tx0 - 6);
    float acc = 0.0f;
#pragma unroll
    for (int ki = 0; ki < 5; ++ki)
#pragma unroll
      for (int kj = 0; kj < 5; ++kj)
        acc += s_g[(by + ki - 2) * GE + (bx + kj - 2)] * GW[ki][kj];
    s_blur[idx] = acc;
  }
  __syncthreads();

  // Stage 4: sobel x/y (reflect), magnitude and quantized phase.
  for (int idx = tid; idx < ME * ME; idx += 256) {
    const int ly = idx / ME, lx = idx - ly * ME;
    const int by = reflect1024(ty0 - 2 + ly) - (ty0 - 4);
    const int bx = reflect1024(tx0 - 2 + lx) - (tx0 - 4);
    float sx = 0.0f, sy = 0.0f;
#pragma unroll
    for (int ki = 0; ki < 5; ++ki)
#pragma unroll
      for (int kj = 0; kj < 5; ++kj) {
        const float v = s_blur[(by + ki - 2) * BE + (bx + kj - 2)];
        sx += v * SW[kj][ki];   // sobel_x_w = s.T
        sy += v * SW[ki][kj];   // sobel_y_w = s
      }
    s_mag[idx] = sqrtf(sx * sx + sy * sy);
    const float ph = atan2f(sx, sy + 1e-5f);
    int q = (int)rintf(ph / 0.7853981633974483f) + 4;  // round-half-even, like jnp.round
    if (q >= 8) q -= 8;
    s_ph[idx] = q;
  }
  __syncthreads();

  // Stage 5: NMS (zero-padded neighbor taps) + low threshold.
  {
    const int d0y[4] = {-1, -1, 0, -1}, d0x[4] = {0, -1, -1, 1};
    const int d1y[4] = { 1,  1, 0,  1}, d1x[4] = {0,  1,  1, -1};
    for (int idx = tid; idx < NE * NE; idx += 256) {
      const int ly = idx / NE, lx = idx - ly * NE;
      const int ry = reflect1024(ty0 - 1 + ly);   // in-image center
      const int rx = reflect1024(tx0 - 1 + lx);
      const int my = ry - (ty0 - 2), mx = rx - (tx0 - 2);
      float m = s_mag[my * ME + mx];
      const int q = s_ph[my * ME + mx] & 3;       // sel ids repeat mod 4
      const int ay = ry + d0y[q], ax = rx + d0x[q];
      const int by = ry + d1y[q], bx = rx + d1x[q];
      const float n0 = (ay >= 0 && ay < 1024 && ax >= 0 && ax < 1024)
                           ? s_mag[(my + d0y[q]) * ME + (mx + d0x[q])] : 0.0f;
      const float n1 = (by >= 0 && by < 1024 && bx >= 0 && bx < 1024)
                           ? s_mag[(my + d1y[q]) * ME + (mx + d1x[q])] : 0.0f;
      if (m < n0 || m < n1) m = 0.0f;   // non-max suppression
      if (m < 0.1f) m = 0.0f;           // THRESH_LO
      s_nms[idx] = m;
    }
  }
  __syncthreads();

  // Stage 6: hysteresis (3x3 reflect) and output.
  for (int idx = tid; idx < TSZ * TSZ; idx += 256) {
    const int ly = idx >> 5, lx = idx & 31;
    const int cy = ly + 1, cx = lx + 1;
    const float m = s_nms[cy * NE + cx];
    const bool high = m > 0.2f;
    const bool weak = (m > 0.1f) && (m < 0.2f);
    int hn = 0;
#pragma unroll
    for (int dy = -1; dy <= 1; ++dy)
#pragma unroll
      for (int dx = -1; dx <= 1; ++dx)
        hn += (s_nms[(cy + dy) * NE + (cx + dx)] > 0.2f) ? 1 : 0;
    const float o = (high || (weak && hn > 0)) ? m : 0.0f;
    out[((size_t)n << 20) + ((size_t)(ty0 + ly) << 10) + (tx0 + lx)] = o;
  }
}

// ------------------------------- Launch ------------------------------------
extern "C" void kernel_launch(void* const* d_in, const int* in_sizes, int n_in,
                              void* d_out, int out_size, void* d_ws, size_t ws_size,
                              hipStream_t stream) {
  (void)in_sizes; (void)n_in; (void)out_size;
  const float* in  = (const float*)d_in[0];
  float*       out = (float*)d_out;

  const size_t grey_bytes = (size_t)16 * NPIX * sizeof(float);   // 64 MB
  const bool   cache_grey = ws_size >= grey_bytes + sizeof(unsigned);

  dim3 grid(1024 / TSZ, 1024 / TSZ, 16);
  if (cache_grey) {
    float*    greyws = (float*)d_ws;
    unsigned* gmx    = (unsigned*)((char*)d_ws + grey_bytes);
    hipMemsetAsync(gmx, 0, sizeof(unsigned), stream);            // 0u == 0.0f
    canny_max_kernel<true><<<4096, 256, 0, stream>>>(in, greyws, gmx);
    canny_main_kernel<true><<<grid, 256, 0, stream>>>(in, greyws, out, gmx);
  } else {
    unsigned* gmx = (unsigned*)d_ws;                             // 4 bytes of scratch
    hipMemsetAsync(gmx, 0, sizeof(unsigned), stream);
    canny_max_kernel<false><<<4096, 256, 0, stream>>>(in, nullptr, gmx);
    canny_main_kernel<false><<<grid, 256, 0, stream>>>(in, nullptr, out, gmx);
  }
}